// LATTE_5325759447087
// MI455X (gfx1250) — compile-verified
//
#include <hip/hip_runtime.h>
#include <math.h>

// ---------------------------------------------------------------------------
// MI455X (gfx1250) implementation of the HAN metapath-attention reference.
//
// Roofline: GEMM h = x@W^T is 3.3 GFLOP / ~100MB -> memory bound, so use
// exact-precision V_WMMA_F32_16X16X4_F32. Edge phase is 3.2M edges x 1KB of
// random gather+atomic traffic; h (51MB) and acc (51MB) both fit in the
// 192MB L2, so L2-resident float atomics are the fast path. beta is folded
// into the scatter weight so only ONE accumulator [N,D] is needed (no rels).
// ---------------------------------------------------------------------------

typedef __attribute__((ext_vector_type(2))) float v2f;
typedef __attribute__((ext_vector_type(8))) float v8f;

#define DIM 128
#define TM  64            // rows of x per workgroup in the GEMM
#define XP  (DIM + 4)     // LDS row pad (stride%64banks=4 -> conflict free)
#define WKC 32            // K-chunk of W staged in LDS
#define WP  (WKC + 4)

// ---------------- GEMM: h = x @ W_lin^T + b_lin  (fp32 WMMA) ---------------
__global__ __launch_bounds__(256) void gemm_h_kernel(
    const float* __restrict__ x, const float* __restrict__ W,
    const float* __restrict__ bias, float* __restrict__ h, int N)
{
    __shared__ float sX[TM][XP];    // 64 x 132 f32 = 33.8 KB
    __shared__ float sW[DIM][WP];   // 128 x 36 f32 = 18.4 KB

    const int t    = threadIdx.x;
    const int row0 = blockIdx.x * TM;

    // Stage x tile [64 x 128] (coalesced float4 loads, scalar LDS stores)
    for (int i = t; i < TM * (DIM / 4); i += 256) {
        int r  = i >> 5;
        int c4 = (i & 31) << 2;
        int gr = row0 + r;
        float4 v = make_float4(0.f, 0.f, 0.f, 0.f);
        if (gr < N) v = *(const float4*)(x + (size_t)gr * DIM + c4);
        sX[r][c4 + 0] = v.x; sX[r][c4 + 1] = v.y;
        sX[r][c4 + 2] = v.z; sX[r][c4 + 3] = v.w;
    }

    const int wave = t >> 5, lane = t & 31;
    const int rt  = wave >> 1;            // row tile 0..3 (16 rows each)
    const int ch  = wave & 1;             // column half (64 cols each)
    const int m16 = lane & 15;
    const int kh  = (lane >> 4) << 1;     // K select: 0 (lanes<16) / 2
    const int cb  = ch * 64 + m16;        // base output column for this lane

    v8f c0 = {}, c1 = {}, c2 = {}, c3 = {};
    const float* ax = &sX[rt * 16 + m16][0];

    for (int kc = 0; kc < DIM; kc += WKC) {
        __syncthreads();  // protect sW reuse (also covers first sX fill)
        // Stage W[:, kc..kc+31]  (B[k][n] = W[n][k], so sW[col][k])
        for (int i = t; i < DIM * (WKC / 4); i += 256) {
            int r  = i >> 3;
            int c4 = (i & 7) << 2;
            float4 v = *(const float4*)(W + (size_t)r * DIM + kc + c4);
            sW[r][c4 + 0] = v.x; sW[r][c4 + 1] = v.y;
            sW[r][c4 + 2] = v.z; sW[r][c4 + 3] = v.w;
        }
        __syncthreads();

        #pragma unroll
        for (int k = 0; k < WKC; k += 4) {
            v2f a;
            a.x = ax[kc + k + kh];
            a.y = ax[kc + k + kh + 1];
            v2f b;
            b.x = sW[cb][k + kh];       b.y = sW[cb][k + kh + 1];
            c0 = __builtin_amdgcn_wmma_f32_16x16x4_f32(false, a, false, b,
                                                       (short)0, c0, false, false);
            b.x = sW[cb + 16][k + kh];  b.y = sW[cb + 16][k + kh + 1];
            c1 = __builtin_amdgcn_wmma_f32_16x16x4_f32(false, a, false, b,
                                                       (short)0, c1, false, false);
            b.x = sW[cb + 32][k + kh];  b.y = sW[cb + 32][k + kh + 1];
            c2 = __builtin_amdgcn_wmma_f32_16x16x4_f32(false, a, false, b,
                                                       (short)0, c2, false, false);
            b.x = sW[cb + 48][k + kh];  b.y = sW[cb + 48][k + kh + 1];
            c3 = __builtin_amdgcn_wmma_f32_16x16x4_f32(false, a, false, b,
                                                       (short)0, c3, false, false);
        }
    }

    // Epilogue: VGPR g holds M=g (lanes 0..15) / M=g+8 (lanes 16..31)
    const int mbase = row0 + rt * 16 + ((lane >> 4) << 3);
    #pragma unroll
    for (int g = 0; g < 8; ++g) {
        int m = mbase + g;
        if (m < N) {
            float* hp = h + (size_t)m * DIM;
            hp[cb     ] = c0[g] + bias[cb     ];
            hp[cb + 16] = c1[g] + bias[cb + 16];
            hp[cb + 32] = c2[g] + bias[cb + 32];
            hp[cb + 48] = c3[g] + bias[cb + 48];
        }
    }
}

// ------------- per-node stats: beta softmax, alpha-scaled al/ar ------------
__global__ __launch_bounds__(256) void row_stats_kernel(
    const float* __restrict__ x, const float* __restrict__ h,
    const float* __restrict__ Wc, const float* __restrict__ bc,
    const float* __restrict__ wl, const float* __restrict__ bl,
    const float* __restrict__ wr, const float* __restrict__ br,
    const float* __restrict__ alpha,
    float* __restrict__ beta, float* __restrict__ alm, float* __restrict__ arm,
    int N)
{
    const int wv   = (int)((blockIdx.x * 256u + threadIdx.x) >> 5);
    const int lane = threadIdx.x & 31;
    if (wv >= N) return;

    const float4 xv = ((const float4*)(x + (size_t)wv * DIM))[lane];
    const float4 hv = ((const float4*)(h + (size_t)wv * DIM))[lane];

    const float* wp[7] = { Wc, Wc + DIM, Wc + 2 * DIM, wl, wl + DIM, wr, wr + DIM };
    float s[7];
    #pragma unroll
    for (int r = 0; r < 7; ++r) {
        const float4 w4 = ((const float4*)(wp[r]))[lane];
        const float4 v  = (r < 3) ? xv : hv;
        s[r] = v.x * w4.x + v.y * w4.y + v.z * w4.z + v.w * w4.w;
        #pragma unroll
        for (int o = 16; o; o >>= 1) s[r] += __shfl_xor(s[r], o, 32);
    }

    if (lane == 0) {
        float l0 = s[0] + bc[0], l1 = s[1] + bc[1], l2 = s[2] + bc[2];
        float mx = fmaxf(l0, fmaxf(l1, l2));
        float e0 = __expf(l0 - mx), e1 = __expf(l1 - mx), e2 = __expf(l2 - mx);
        float inv = 1.0f / (e0 + e1 + e2);
        beta[wv * 3 + 0] = e0 * inv;
        beta[wv * 3 + 1] = e1 * inv;
        beta[wv * 3 + 2] = e2 * inv;
        alm[wv]     = alpha[0] * (s[3] + bl[0]);
        alm[N + wv] = alpha[1] * (s[4] + bl[1]);
        arm[wv]     = alpha[0] * (s[5] + br[0]);
        arm[N + wv] = alpha[1] * (s[6] + br[1]);
    }
}

// ------------------------------ init scratch -------------------------------
__global__ void init_kernel(float* __restrict__ acc, float* __restrict__ denom,
                            unsigned* __restrict__ maxk, long long accN, int twoN)
{
    const long long i0 = (long long)blockIdx.x * 256 + threadIdx.x;
    const long long st = (long long)gridDim.x * 256;
    for (long long i = i0; i < accN; i += st) acc[i] = 0.f;
    for (long long i = i0; i < twoN; i += st) { denom[i] = 0.f; maxk[i] = 0u; }
}

// monotone uint encoding for float atomic max
__device__ __forceinline__ unsigned fenc(float f) {
    unsigned u = __float_as_uint(f);
    return (u & 0x80000000u) ? ~u : (u | 0x80000000u);
}
__device__ __forceinline__ float fdec(unsigned k) {
    return __uint_as_float((k & 0x80000000u) ? (k & 0x7FFFFFFFu) : ~k);
}

// ------------- pass 1: logits + per-dst segment max (atomic) ---------------
__global__ __launch_bounds__(256) void edge_logit_max_kernel(
    const int* __restrict__ ei, const float* __restrict__ alm,
    const float* __restrict__ arm, float* __restrict__ logit,
    unsigned* __restrict__ maxk, int E)
{
    const int e = (int)(blockIdx.x * 256u + threadIdx.x);
    if (e >= E) return;
    const int dst = ei[e];
    const int src = ei[E + e];
    const float lg = alm[src] + arm[dst];
    logit[e] = lg;
    atomicMax(maxk + dst, fenc(lg));
}

// ------------- pass 2: exp(logit - max), per-dst denominator ---------------
__global__ __launch_bounds__(256) void edge_exp_sum_kernel(
    const int* __restrict__ ei, float* __restrict__ logit,
    const unsigned* __restrict__ maxk, float* __restrict__ denom, int E)
{
    const int e = (int)(blockIdx.x * 256u + threadIdx.x);
    if (e >= E) return;
    const int dst = ei[e];
    const float mx = fdec(maxk[dst]);
    const float ex = __expf(logit[e] - mx);
    logit[e] = ex;   // reuse buffer as exp storage
    atomicAdd(denom + dst, ex);
}

// ---- pass 3: scatter beta[dst,m] * softmax * h[src] into shared acc -------
__global__ __launch_bounds__(256) void edge_scatter_kernel(
    const int* __restrict__ ei, const float* __restrict__ exbuf,
    const float* __restrict__ denom, const float* __restrict__ beta,
    const float* __restrict__ h, float* __restrict__ acc, int E, int m)
{
    const int wv   = (int)((blockIdx.x * 256u + threadIdx.x) >> 5);
    const int lane = threadIdx.x & 31;
    if (wv >= E) return;
    const int dst = ei[wv];
    const int src = ei[E + wv];
    const float w = beta[dst * 3 + m] * exbuf[wv] / (denom[dst] + 1e-16f);
    const float4 hv = ((const float4*)(h + (size_t)src * DIM))[lane];
    float* a = acc + (size_t)dst * DIM + (lane << 2);
    atomicAdd(a + 0, hv.x * w);
    atomicAdd(a + 1, hv.y * w);
    atomicAdd(a + 2, hv.z * w);
    atomicAdd(a + 3, hv.w * w);
}

// ------------------- out = relu(acc + beta[:,2] * h) -----------------------
__global__ __launch_bounds__(256) void final_kernel(
    const float* __restrict__ acc, const float* __restrict__ h,
    const float* __restrict__ beta, float* __restrict__ out, int N)
{
    const int i = (int)(blockIdx.x * 256u + threadIdx.x);  // float4 index
    if (i >= N * (DIM / 4)) return;
    const int n = i >> 5;
    const float bs = beta[n * 3 + 2];
    const float4 a  = ((const float4*)acc)[i];
    const float4 hh = ((const float4*)h)[i];
    float4 o;
    o.x = fmaxf(fmaf(bs, hh.x, a.x), 0.f);
    o.y = fmaxf(fmaf(bs, hh.y, a.y), 0.f);
    o.z = fmaxf(fmaf(bs, hh.z, a.z), 0.f);
    o.w = fmaxf(fmaf(bs, hh.w, a.w), 0.f);
    ((float4*)out)[i] = o;
}

// ---------------------------------------------------------------------------
extern "C" void kernel_launch(void* const* d_in, const int* in_sizes, int n_in,
                              void* d_out, int out_size, void* d_ws, size_t ws_size,
                              hipStream_t stream)
{
    const float* x    = (const float*)d_in[0];
    const int*   ei1  = (const int*)  d_in[1];
    const int*   ei2  = (const int*)  d_in[2];
    const float* Wl   = (const float*)d_in[3];
    const float* blin = (const float*)d_in[4];
    const float* Wc   = (const float*)d_in[5];
    const float* bc   = (const float*)d_in[6];
    const float* alw  = (const float*)d_in[7];
    const float* alb  = (const float*)d_in[8];
    const float* arw  = (const float*)d_in[9];
    const float* arb  = (const float*)d_in[10];
    const float* alph = (const float*)d_in[11];

    const int N  = in_sizes[0] / DIM;
    const int Ea = in_sizes[1] / 2;
    const int Eb = in_sizes[2] / 2;
    float* out = (float*)d_out;

    // workspace carve-out (256B aligned)
    char* ws = (char*)d_ws;
    auto carve = [&](size_t bytes) -> char* {
        char* p = ws;
        ws += (bytes + 255) & ~(size_t)255;
        return p;
    };
    float*    h     = (float*)   carve((size_t)N * DIM * 4);
    float*    acc   = (float*)   carve((size_t)N * DIM * 4);
    float*    beta  = (float*)   carve((size_t)N * 3 * 4);
    float*    alm   = (float*)   carve((size_t)N * 2 * 4);
    float*    arm   = (float*)   carve((size_t)N * 2 * 4);
    unsigned* maxk  = (unsigned*)carve((size_t)N * 2 * 4);
    float*    denom = (float*)   carve((size_t)N * 2 * 4);
    const size_t Emax = (size_t)(Ea > Eb ? Ea : Eb);
    float*    exbuf = (float*)   carve(Emax * 4);

    init_kernel<<<4096, 256, 0, stream>>>(acc, denom, maxk,
                                          (long long)N * DIM, 2 * N);
    gemm_h_kernel<<<(N + TM - 1) / TM, 256, 0, stream>>>(x, Wl, blin, h, N);
    row_stats_kernel<<<(N + 7) / 8, 256, 0, stream>>>(
        x, h, Wc, bc, alw, alb, arw, arb, alph, beta, alm, arm, N);

    const int* eis[2] = { ei1, ei2 };
    const int  Es[2]  = { Ea, Eb };
    for (int m = 0; m < 2; ++m) {
        const int* ei = eis[m];
        const int  E  = Es[m];
        edge_logit_max_kernel<<<(E + 255) / 256, 256, 0, stream>>>(
            ei, alm + (size_t)m * N, arm + (size_t)m * N, exbuf,
            maxk + (size_t)m * N, E);
        edge_exp_sum_kernel<<<(E + 255) / 256, 256, 0, stream>>>(
            ei, exbuf, maxk + (size_t)m * N, denom + (size_t)m * N, E);
        const int nblk = (int)(((size_t)E * 32 + 255) / 256);
        edge_scatter_kernel<<<nblk, 256, 0, stream>>>(
            ei, exbuf, denom + (size_t)m * N, beta, h, acc, E, m);
    }
    final_kernel<<<(N * (DIM / 4) + 255) / 256, 256, 0, stream>>>(
        acc, h, beta, out, N);
}